// Tokenizer_40458591929010
// MI455X (gfx1250) — compile-verified
//
#include <hip/hip_runtime.h>

typedef float v2f __attribute__((ext_vector_type(2)));
typedef float v4f __attribute__((ext_vector_type(4)));
typedef float v8f __attribute__((ext_vector_type(8)));

#define VOCAB 4096
#define EMBED 64
#define NUM_TOKENS 16
#define OBS_DIM 128
#define HID 512
#define BN 4096                    // 1024 * 4 rows
#define ZROWS (BN * NUM_TOKENS)    // 65536 embedding rows
#define LDS_PITCH 68               // 64 + 4 dword pad -> conflict-free b64 reads

// fp32 WMMA: D(16x16,f32) = A(16x4,f32) x B(4x16,f32) + C
__device__ __forceinline__ v8f wmma_f32(v2f a, v2f b, v8f c) {
    return __builtin_amdgcn_wmma_f32_16x16x4_f32(
        false, a, false, b, (short)0, c, false, false);
}

// ---------------------------------------------------------------------------
// ||codebook[j]||^2 for all 4096 codes
// ---------------------------------------------------------------------------
__global__ __launch_bounds__(256) void cnorm_kernel(const float* __restrict__ cb,
                                                    float* __restrict__ cnorm) {
    const int j = blockIdx.x * blockDim.x + threadIdx.x;   // exact: 4096 threads
    const v4f* row = (const v4f*)(cb + (size_t)j * EMBED);
    float s = 0.0f;
#pragma unroll
    for (int q = 0; q < EMBED / 4; ++q) {
        v4f v = row[q];
        s += v.x * v.x + v.y * v.y + v.z * v.z + v.w * v.w;
    }
    cnorm[j] = s;
}

// ---------------------------------------------------------------------------
// C = act(A(MxK) @ W(KxN) + bias), row-major, f32 WMMA 16x16x4.
// One wave computes a 16x64 tile (4 accumulators; A fragment reused 4x).
// ---------------------------------------------------------------------------
template <int RELU>
__global__ __launch_bounds__(128) void gemm_bias_kernel(
    const float* __restrict__ A, const float* __restrict__ W,
    const float* __restrict__ bias, float* __restrict__ C,
    int M, int N, int K) {
    const int lane = threadIdx.x & 31;
    const int wave = blockIdx.x * 4 + (threadIdx.x >> 5);
    const int nGroups = N >> 6;
    const int m0 = (wave / nGroups) << 4;
    const int n0 = (wave % nGroups) << 6;
    const int i = lane & 15;
    const int h = lane >> 4;

    v8f acc0 = {}, acc1 = {}, acc2 = {}, acc3 = {};

    const float* arow = A + (size_t)(m0 + i) * K + (h << 1);
    for (int kc = 0; kc < (K >> 2); ++kc) {
        const int k = (kc << 2) + (h << 1);
        v2f a = *(const v2f*)(arow + (kc << 2));
        const float* w0 = W + (size_t)k * N + n0 + i;
        const float* w1 = w0 + N;
        v2f b;
        b.x = w0[0];  b.y = w1[0];  acc0 = wmma_f32(a, b, acc0);
        b.x = w0[16]; b.y = w1[16]; acc1 = wmma_f32(a, b, acc1);
        b.x = w0[32]; b.y = w1[32]; acc2 = wmma_f32(a, b, acc2);
        b.x = w0[48]; b.y = w1[48]; acc3 = wmma_f32(a, b, acc3);
    }

    const float b0 = bias[n0 + i];
    const float b1 = bias[n0 + 16 + i];
    const float b2 = bias[n0 + 32 + i];
    const float b3 = bias[n0 + 48 + i];
#pragma unroll
    for (int v = 0; v < 8; ++v) {
        const int row = m0 + v + (h << 3);
        float* crow = C + (size_t)row * N + n0 + i;
        float x0 = acc0[v] + b0, x1 = acc1[v] + b1;
        float x2 = acc2[v] + b2, x3 = acc3[v] + b3;
        if (RELU) {
            x0 = fmaxf(x0, 0.0f); x1 = fmaxf(x1, 0.0f);
            x2 = fmaxf(x2, 0.0f); x3 = fmaxf(x3, 0.0f);
        }
        crow[0] = x0; crow[16] = x1; crow[32] = x2; crow[48] = x3;
    }
}

// ---------------------------------------------------------------------------
// Fused VQ. Block = 128 threads = 4 waves, each wave owns 32 zf rows.
// Codebook tiles (16 codes x 64 f32) staged in LDS, double-buffered and
// shared by all 4 waves; each B fragment feeds 2 WMMAs.
// argmin_j (||c_j||^2 - 2 z.c_j)   (||z||^2 drops out of the argmin),
// then gather codebook[token] -> z_q (== decoder input, straight-through).
// ---------------------------------------------------------------------------
__global__ __launch_bounds__(128) void vq_kernel(
    const float* __restrict__ zf, const float* __restrict__ cb,
    const float* __restrict__ cnorm, float* __restrict__ zq) {
    const int lane = threadIdx.x & 31;
    const int w = threadIdx.x >> 5;
    const int i = lane & 15;
    const int h = lane >> 4;
    const int m0 = blockIdx.x * 128 + w * 32;

    __shared__ float bt[2][16][LDS_PITCH];

    // A fragments: 32 rows x 64 K, two register sets of 16 K-chunks.
    v2f a0[16], a1[16];
    const float* zr0 = zf + (size_t)(m0 + i) * EMBED + (h << 1);
    const float* zr1 = zf + (size_t)(m0 + 16 + i) * EMBED + (h << 1);
#pragma unroll
    for (int kc = 0; kc < 16; ++kc) {
        a0[kc] = *(const v2f*)(zr0 + (kc << 2));
        a1[kc] = *(const v2f*)(zr1 + (kc << 2));
    }

    // Cooperative tile fill mapping: 8 threads per code row, 8 floats each.
    const int lr = threadIdx.x >> 3;        // code row 0..15
    const int lc = (threadIdx.x & 7) << 3;  // col 0..56
    {
        const float* src = cb + (size_t)lr * EMBED + lc;
        *(v4f*)(&bt[0][lr][lc])     = *(const v4f*)(src);
        *(v4f*)(&bt[0][lr][lc + 4]) = *(const v4f*)(src + 4);
    }
    __syncthreads();

    float best0[8], best1[8];
    int bi0[8], bi1[8];
#pragma unroll
    for (int v = 0; v < 8; ++v) {
        best0[v] = 3.4e38f; best1[v] = 3.4e38f;
        bi0[v] = 0; bi1[v] = 0;
    }

    for (int t = 0; t < VOCAB / 16; ++t) {
        const int n0 = t << 4;
        const int cur = t & 1;
        if (t + 1 < VOCAB / 16) {              // uniform branch: EXEC stays full
            const float* src = cb + (size_t)(n0 + 16 + lr) * EMBED + lc;
            v4f p0 = *(const v4f*)(src);
            v4f p1 = *(const v4f*)(src + 4);
            *(v4f*)(&bt[cur ^ 1][lr][lc])     = p0;
            *(v4f*)(&bt[cur ^ 1][lr][lc + 4]) = p1;
        }
        v8f acc0 = {}, acc1 = {};
        const float* brow = &bt[cur][i][h << 1];   // banks 4i+2h: conflict-free
#pragma unroll
        for (int kc = 0; kc < 16; ++kc) {
            v2f b = *(const v2f*)(brow + (kc << 2));
            acc0 = wmma_f32(a0[kc], b, acc0);
            acc1 = wmma_f32(a1[kc], b, acc1);
        }
        const float cn = cnorm[n0 + i];
#pragma unroll
        for (int v = 0; v < 8; ++v) {
            const float s0 = cn - 2.0f * acc0[v];
            if (s0 < best0[v]) { best0[v] = s0; bi0[v] = n0 + i; }
            const float s1 = cn - 2.0f * acc1[v];
            if (s1 < best1[v]) { best1[v] = s1; bi1[v] = n0 + i; }
        }
        __syncthreads();
    }

    // Per-row min+index reduction across the 16 lanes of each half, then
    // gather the winning code row into z_q with float4 stores.
#pragma unroll
    for (int v = 0; v < 8; ++v) {
        float bv = best0[v];
        int bi = bi0[v];
#pragma unroll
        for (int mask = 8; mask >= 1; mask >>= 1) {
            const float ov = __shfl_xor(bv, mask, 16);
            const int oi = __shfl_xor(bi, mask, 16);
            if (ov < bv || (ov == bv && oi < bi)) { bv = ov; bi = oi; }
        }
        const int row = m0 + v + (h << 3);
        *(v4f*)(zq + (size_t)row * EMBED + (i << 2)) =
            *(const v4f*)(cb + (size_t)bi * EMBED + (i << 2));
    }
#pragma unroll
    for (int v = 0; v < 8; ++v) {
        float bv = best1[v];
        int bi = bi1[v];
#pragma unroll
        for (int mask = 8; mask >= 1; mask >>= 1) {
            const float ov = __shfl_xor(bv, mask, 16);
            const int oi = __shfl_xor(bi, mask, 16);
            if (ov < bv || (ov == bv && oi < bi)) { bv = ov; bi = oi; }
        }
        const int row = m0 + 16 + v + (h << 3);
        *(v4f*)(zq + (size_t)row * EMBED + (i << 2)) =
            *(const v4f*)(cb + (size_t)bi * EMBED + (i << 2));
    }
}

// ---------------------------------------------------------------------------
extern "C" void kernel_launch(void* const* d_in, const int* in_sizes, int n_in,
                              void* d_out, int out_size, void* d_ws, size_t ws_size,
                              hipStream_t stream) {
    const float* obs = (const float*)d_in[0];
    const float* ew1 = (const float*)d_in[1];
    const float* eb1 = (const float*)d_in[2];
    const float* ew2 = (const float*)d_in[3];
    const float* eb2 = (const float*)d_in[4];
    const float* dw1 = (const float*)d_in[5];
    const float* db1 = (const float*)d_in[6];
    const float* dw2 = (const float*)d_in[7];
    const float* db2 = (const float*)d_in[8];
    const float* cb  = (const float*)d_in[9];

    float* out_z   = (float*)d_out;                        // (4096, 1024)
    float* out_zq  = out_z + (size_t)BN * (NUM_TOKENS * EMBED);
    float* out_rec = out_zq + (size_t)BN * (NUM_TOKENS * EMBED);

    float* ws_h  = (float*)d_ws;                           // (4096, 512), reused
    float* ws_cn = ws_h + (size_t)BN * HID;                // (4096,)

    cnorm_kernel<<<VOCAB / 256, 256, 0, stream>>>(cb, ws_cn);

    // H = relu(obs @ enc_w1 + enc_b1)   M=4096 K=128 N=512
    gemm_bias_kernel<1><<<(BN / 16) * (HID / 64) / 4, 128, 0, stream>>>(
        obs, ew1, eb1, ws_h, BN, HID, OBS_DIM);

    // Z = H @ enc_w2 + enc_b2           M=4096 K=512 N=1024  -> z output
    gemm_bias_kernel<0><<<(BN / 16) * ((NUM_TOKENS * EMBED) / 64) / 4, 128, 0, stream>>>(
        ws_h, ew2, eb2, out_z, BN, NUM_TOKENS * EMBED, HID);

    // VQ: argmin + gather -> z_q output (also the decoder input)
    vq_kernel<<<ZROWS / 128, 128, 0, stream>>>(out_z, cb, ws_cn, out_zq);

    // D1 = relu(z_q @ dec_w1 + dec_b1)  M=4096 K=1024 N=512
    gemm_bias_kernel<1><<<(BN / 16) * (HID / 64) / 4, 128, 0, stream>>>(
        out_zq, dw1, db1, ws_h, BN, HID, NUM_TOKENS * EMBED);

    // rec = D1 @ dec_w2 + dec_b2        M=4096 K=512 N=128  -> rec output
    gemm_bias_kernel<0><<<(BN / 16) * (OBS_DIM / 64) / 4, 128, 0, stream>>>(
        ws_h, dw2, db2, out_rec, BN, OBS_DIM, HID);
}